// TritonExpertDispatch_1322849927639
// MI455X (gfx1250) — compile-verified
//
#include <hip/hip_runtime.h>
#include <hip/hip_bf16.h>
#include <stdint.h>

// ---------------------------------------------------------------------------
// MoE expert dispatch (gather/scatter), MI455X (gfx1250).
// Memory-bound: ~128MB gather read + ~160MB scattered write => ~12.5us floor
// at 23.3 TB/s. No matmul -> no WMMA; the CDNA5 feature used is the async
// global<->LDS DMA path (ASYNCcnt) for the 8KB row copies.
// ---------------------------------------------------------------------------

#define NUM_EXPERTS 64
#define NUM_TOKENS  16384
#define EMBED_DIM   2048
#define CAPACITY    320
#define TOK_PER_BLK 256
#define NUM_BLKS    (NUM_TOKENS / TOK_PER_BLK)   // 64
#define NUM_SLOTS   (NUM_EXPERTS * CAPACITY)     // 20480

#if __has_builtin(__builtin_amdgcn_global_load_async_to_lds_b128) && \
    __has_builtin(__builtin_amdgcn_global_store_async_from_lds_b128)
#define HAVE_ASYNC_LDS 1
#else
#define HAVE_ASYNC_LDS 0
#endif

#if __has_builtin(__builtin_amdgcn_s_wait_asynccnt)
#define WAIT_ASYNC0() __builtin_amdgcn_s_wait_asynccnt(0)
#else
#define WAIT_ASYNC0() asm volatile("s_wait_asynccnt 0" ::: "memory")
#endif

// Builtin signature (from hipcc diagnostic): pointee is a 16-byte int vector,
// global side in addrspace(1), LDS side in addrspace(3).
typedef int v4i __attribute__((ext_vector_type(4)));
typedef __attribute__((address_space(1))) v4i* gptr_v4i;
typedef __attribute__((address_space(3))) v4i* lptr_v4i;

#define GAS(p) ((gptr_v4i)(p))
#define LAS(p) ((lptr_v4i)(p))

// ---------------------------------------------------------------------------
// Kernel 0: per-call re-init of slot map + small outputs (harness poisons
// once and never re-poisons; we must regenerate defaults every call).
// ---------------------------------------------------------------------------
__global__ void init_kernel(int* __restrict__ slot_token,
                            float* __restrict__ out_w,
                            float* __restrict__ out_ti,
                            int* __restrict__ dropped) {
    int i = blockIdx.x * blockDim.x + threadIdx.x;
    if (i < NUM_SLOTS) {
        slot_token[i] = -1;
        out_w[i]  = 0.0f;
        out_ti[i] = -1.0f;
    }
    if (i == 0) dropped[0] = 0;
}

// ---------------------------------------------------------------------------
// Kernel 1: per-block expert histogram (LDS atomics; order-independent).
// ---------------------------------------------------------------------------
__global__ __launch_bounds__(TOK_PER_BLK)
void hist_kernel(const int* __restrict__ eids, int* __restrict__ counts) {
    __shared__ int h[NUM_EXPERTS];
    int tid = threadIdx.x, b = blockIdx.x;
    if (tid < NUM_EXPERTS) h[tid] = 0;
    __syncthreads();
    int e = eids[b * TOK_PER_BLK + tid] & (NUM_EXPERTS - 1);
    atomicAdd(&h[e], 1);
    __syncthreads();
    if (tid < NUM_EXPERTS) counts[b * NUM_EXPERTS + tid] = h[tid];
}

// ---------------------------------------------------------------------------
// Kernel 2: per-expert exclusive prefix over blocks (counting-sort offsets).
// Tiny (64x64 ints), one block of 64 threads, serial over 64 blocks.
// ---------------------------------------------------------------------------
__global__ void prefix_kernel(const int* __restrict__ counts,
                              int* __restrict__ offsets) {
    int e = threadIdx.x;           // 64 threads, one per expert
    int run = 0;
    for (int b = 0; b < NUM_BLKS; ++b) {
        offsets[b * NUM_EXPERTS + e] = run;
        run += counts[b * NUM_EXPERTS + e];
    }
}

// ---------------------------------------------------------------------------
// Kernel 3: stable in-block rank (token order) + scatter of slot map,
// combine weights, token indices; count drops. Rank scan is O(tid) over LDS
// — 64 blocks * 256^2/2 LDS reads, negligible vs 290MB of HBM traffic,
// and exactly reproduces the reference's stable counting sort.
// ---------------------------------------------------------------------------
__global__ __launch_bounds__(TOK_PER_BLK)
void scatter_kernel(const int* __restrict__ eids,
                    const float* __restrict__ w,
                    const int* __restrict__ offsets,
                    int* __restrict__ slot_token,
                    float* __restrict__ out_w,
                    float* __restrict__ out_ti,
                    int* __restrict__ dropped) {
    __shared__ int sh_e[TOK_PER_BLK];
    int tid = threadIdx.x, b = blockIdx.x;
    int i = b * TOK_PER_BLK + tid;
    int e = eids[i] & (NUM_EXPERTS - 1);
    sh_e[tid] = e;
    __syncthreads();
    int rank = 0;
    for (int j = 0; j < tid; ++j) rank += (sh_e[j] == e) ? 1 : 0;
    int pos = offsets[b * NUM_EXPERTS + e] + rank;
    if (pos < CAPACITY) {
        int s = e * CAPACITY + pos;
        slot_token[s] = i;
        out_w[s]  = w[i];
        out_ti[s] = (float)i;
    } else {
        atomicAdd(dropped, 1);
    }
}

// ---------------------------------------------------------------------------
// Kernel 4: one block per (expert,capacity) slot. Valid slot: stream the
// 8KB row x[t] -> LDS -> dispatched_x[slot] via gfx1250 async global<->LDS
// DMA (ASYNCcnt). Empty slot: write zeros (b128 stores). Each lane owns
// disjoint 16B chunks of LDS, EXEC is full, so a wave-local
// s_wait_asynccnt 0 between the loads and the stores is sufficient.
// ---------------------------------------------------------------------------
__global__ __launch_bounds__(128)
void row_copy_kernel(const float* __restrict__ x,
                     const int* __restrict__ slot_token,
                     const int* __restrict__ dropped,
                     float* __restrict__ out_x,
                     float* __restrict__ out_drop) {
    __shared__ float lds_buf[EMBED_DIM];       // 8KB staging buffer
    int slot = blockIdx.x;
    int tid  = threadIdx.x;
    int t = slot_token[slot];
    float* dst = out_x + (size_t)slot * EMBED_DIM;

    if (t >= 0) {
        const float* src = x + (size_t)t * EMBED_DIM;
#if HAVE_ASYNC_LDS
        // 4 x (128 lanes * 16B) = 8KB row
        #pragma unroll
        for (int k = 0; k < 4; ++k) {
            int off = (tid + k * 128) * 4;     // float index, 16B aligned
            __builtin_amdgcn_global_load_async_to_lds_b128(
                GAS(src + off), LAS(lds_buf + off), 0, 0);
        }
        WAIT_ASYNC0();                          // LDS now holds the row
        #pragma unroll
        for (int k = 0; k < 4; ++k) {
            int off = (tid + k * 128) * 4;
            __builtin_amdgcn_global_store_async_from_lds_b128(
                GAS(dst + off), LAS(lds_buf + off), 0, 0);
        }
        // s_endpgm implies wait-idle; no explicit final wait needed.
#else
        #pragma unroll
        for (int k = 0; k < 4; ++k) {
            int off = (tid + k * 128) * 4;
            *(float4*)(dst + off) = *(const float4*)(src + off);
        }
#endif
    } else {
        float4 z; z.x = 0.f; z.y = 0.f; z.z = 0.f; z.w = 0.f;
        #pragma unroll
        for (int k = 0; k < 4; ++k) {
            int off = (tid + k * 128) * 4;
            *(float4*)(dst + off) = z;
        }
    }

    // Final scalar output: tokens_dropped (stream-ordered after scatter).
    if (slot == 0 && tid == 0) out_drop[0] = (float)dropped[0];
}

// ---------------------------------------------------------------------------
// Launch
// ---------------------------------------------------------------------------
extern "C" void kernel_launch(void* const* d_in, const int* in_sizes, int n_in,
                              void* d_out, int out_size, void* d_ws, size_t ws_size,
                              hipStream_t stream) {
    (void)in_sizes; (void)n_in; (void)out_size; (void)ws_size;

    const float* x    = (const float*)d_in[0];
    const int*   eids = (const int*)d_in[1];     // integer inputs -> int32
    const float* w    = (const float*)d_in[2];

    // d_out: [dispatched_x | combine_weights | token_indices | tokens_dropped]
    float* out_x    = (float*)d_out;
    float* out_w    = out_x + (size_t)NUM_SLOTS * EMBED_DIM;
    float* out_ti   = out_w + NUM_SLOTS;
    float* out_drop = out_ti + NUM_SLOTS;

    // d_ws: counts[64][64] | offsets[64][64] | slot_token[20480] | dropped[1]
    int* counts     = (int*)d_ws;
    int* offsets    = counts  + NUM_BLKS * NUM_EXPERTS;
    int* slot_token = offsets + NUM_BLKS * NUM_EXPERTS;
    int* dropped    = slot_token + NUM_SLOTS;

    init_kernel<<<(NUM_SLOTS + 255) / 256, 256, 0, stream>>>(
        slot_token, out_w, out_ti, dropped);

    hist_kernel<<<NUM_BLKS, TOK_PER_BLK, 0, stream>>>(eids, counts);

    prefix_kernel<<<1, NUM_EXPERTS, 0, stream>>>(counts, offsets);

    scatter_kernel<<<NUM_BLKS, TOK_PER_BLK, 0, stream>>>(
        eids, w, offsets, slot_token, out_w, out_ti, dropped);

    row_copy_kernel<<<NUM_SLOTS, 128, 0, stream>>>(
        x, slot_token, dropped, out_x, out_drop);
}